// GaussianKernels_22763326669354
// MI455X (gfx1250) — compile-verified
//
#include <hip/hip_runtime.h>
#include <math.h>
#include <stdint.h>

// ---- problem constants (match reference) ----
#define BROWS   2048
#define DDIM    128
#define NCENT   100000
#define KNN     200u
#define NCLS    1000
#define GCONST  0.005f        // 1/(2*sigma^2), sigma=10
// ---- tiling ----
#define HBINS        1024
#define TILE_N       64
#define TILE_STRIDE  132      // 128 + 4 pad -> conflict-free b64 LDS reads
#define TILE_FLOATS  (TILE_N * TILE_STRIDE)
#define GTHREADS     128      // 4 waves of 32
#define CSPLIT       16

// ---- gfx1250 async global->LDS path (with portable fallback) ----
#if defined(__has_builtin)
#if __has_builtin(__builtin_amdgcn_global_load_async_to_lds_b128)
#define HAS_ASYNC_LDS 1
#endif
#endif
#ifndef HAS_ASYNC_LDS
#define HAS_ASYNC_LDS 0
#endif

#if HAS_ASYNC_LDS
#if __has_builtin(__builtin_amdgcn_s_wait_asynccnt)
#define WAIT_ASYNC0() __builtin_amdgcn_s_wait_asynccnt(0)
#else
#define WAIT_ASYNC0() asm volatile("s_wait_asynccnt 0" ::: "memory")
#endif
#else
#define WAIT_ASYNC0() ((void)0)
#endif

typedef float v2f __attribute__((ext_vector_type(2)));
typedef float v8f __attribute__((ext_vector_type(8)));
typedef int   v4i __attribute__((ext_vector_type(4)));
typedef __attribute__((address_space(1))) v4i* gptr_b128;
typedef __attribute__((address_space(3))) v4i* lptr_b128;

// ---------------- squared row norms: one wave per row ----------------
__global__ void sqnorm_kernel(const float* __restrict__ x, float* __restrict__ out, int nrows) {
    int row  = blockIdx.x * (blockDim.x >> 5) + (threadIdx.x >> 5);
    int lane = threadIdx.x & 31;
    if (row >= nrows) return;
    const float4* p = (const float4*)(x + (size_t)row * DDIM);
    float4 a = p[lane];                       // 32 lanes * 4 = 128 elements
    float s = a.x*a.x + a.y*a.y + a.z*a.z + a.w*a.w;
    for (int off = 16; off > 0; off >>= 1) s += __shfl_down(s, off, 32);
    if (lane == 0) out[row] = s;
}

// ---------------- generic zero ----------------
__global__ void zero_kernel(unsigned* __restrict__ p, size_t n) {
    size_t i = (size_t)blockIdx.x * blockDim.x + threadIdx.x;
    size_t stride = (size_t)gridDim.x * blockDim.x;
    for (; i < n; i += stride) p[i] = 0u;
}

// ---------------- tile staging: 64 centres x 128 f32 -> LDS ----------------
__device__ __forceinline__ void issue_tile(const float* __restrict__ centres,
                                           int base, float* __restrict__ dst_tile,
                                           int tid) {
    const int r = tid >> 1, h = (tid & 1) * 64;
    const int gr = base + r;
    if (gr >= NCENT) return;
    const float4* src = (const float4*)(centres + (size_t)gr * DDIM + h);
    float4*       dst = (float4*)(dst_tile + r * TILE_STRIDE + h);
#if HAS_ASYNC_LDS
    #pragma unroll
    for (int j = 0; j < 16; ++j)
        __builtin_amdgcn_global_load_async_to_lds_b128(
            (gptr_b128)(uintptr_t)(src + j),
            (lptr_b128)(unsigned)(uintptr_t)(dst + j),
            0, 0);
#else
    float4 tmp[16];
    #pragma unroll
    for (int j = 0; j < 16; ++j) tmp[j] = src[j];
    #pragma unroll
    for (int j = 0; j < 16; ++j) dst[j] = tmp[j];
#endif
}

// ---------------- fused GEMM passes ----------------
// MODE 0: coarse histogram of float-bit keys (bits 31..21)
// MODE 1: refine histogram (bits 20..11) inside per-row boundary bin
// MODE 2: accumulate Gaussian weights into per-row class bins for d2 <= threshold
template <int MODE>
__global__ void gemm_pass_kernel(const float*  __restrict__ feats,
                                 const float*  __restrict__ centres,
                                 const float*  __restrict__ fn2,
                                 const float*  __restrict__ cn2,
                                 unsigned*     __restrict__ ghist,
                                 const unsigned* __restrict__ binfo,
                                 const unsigned* __restrict__ thresh,
                                 const float*  __restrict__ weight,
                                 const int*    __restrict__ labels,
                                 float*        __restrict__ pacc,
                                 int colChunk) {
    extern __shared__ float smem[];
    float*    bufs  = smem;                                  // 2 x TILE_FLOATS
    unsigned* shist = (unsigned*)(smem + 2 * TILE_FLOATS);   // 16*HBINS (mode 0/1)
    float*    scls  = smem + 2 * TILE_FLOATS;                // 16*NCLS  (mode 2)

    const int tid  = threadIdx.x;          // 0..127
    const int lane = tid & 31;
    const int wv   = tid >> 5;             // wave id 0..3
    const int rb   = blockIdx.x * 16;      // feature-row base
    const int c0   = blockIdx.y * colChunk;
    const int c1   = min(NCENT, c0 + colChunk);

    // zero LDS accumulators (made visible by the first loop barrier)
    const int auxN = (MODE == 2) ? 16 * NCLS : 16 * HBINS;
    if (MODE == 2) { for (int i = tid; i < auxN; i += GTHREADS) scls[i]  = 0.0f; }
    else           { for (int i = tid; i < auxN; i += GTHREADS) shist[i] = 0u;  }

    // --- A fragments, register resident: 16 rows x 128 of features ---
    // lane l holds A[l%16][4t + 2*(l/16) + {0,1}] for t = 0..31
    const int m  = lane & 15;
    const int kh = (lane >> 4) * 2;
    v2f a[32];
    {
        const float* frow = feats + (size_t)(rb + m) * DDIM + kh;
        #pragma unroll
        for (int t = 0; t < 32; ++t) a[t] = *(const v2f*)(frow + 4 * t);
    }
    // per-output-row constants for this lane's 8 accumulator rows
    float    fn2v[8];
    unsigned b1v[8];
    unsigned Tv[8];
    #pragma unroll
    for (int v = 0; v < 8; ++v) {
        const int orow = rb + v + 8 * (lane >> 4);
        fn2v[v] = fn2[orow];
        if (MODE == 1) b1v[v] = binfo[orow * 2];
        if (MODE == 2) Tv[v]  = thresh[orow];
    }

    const int ntiles = (c1 - c0 + TILE_N - 1) / TILE_N;
    issue_tile(centres, c0, bufs, tid);                      // prefetch tile 0

    for (int it = 0; it < ntiles; ++it) {
        const int base = c0 + it * TILE_N;
        float* cur = bufs + (it & 1) * TILE_FLOATS;
        float* nxt = bufs + ((it + 1) & 1) * TILE_FLOATS;

        WAIT_ASYNC0();          // current tile's async writes landed in LDS
        __syncthreads();        // everyone done reading `nxt`'s previous contents
        if (it + 1 < ntiles) issue_tile(centres, base + TILE_N, nxt, tid);

        // each wave computes a 16x16 f32 tile: 32 x V_WMMA_F32_16X16X4_F32
        const int nl = wv * 16 + (lane & 15);   // local column
        const int n  = base + nl;               // global centre index
        const float* bp = cur + nl * TILE_STRIDE + kh;
        v8f acc = {};
        #pragma unroll
        for (int t = 0; t < 32; ++t) {
            v2f b = *(const v2f*)(bp + 4 * t);
            acc = __builtin_amdgcn_wmma_f32_16x16x4_f32(
                      false, a[t], false, b, (short)0, acc, false, false);
        }

        const bool  valid = (n < NCENT);
        const float c2    = valid ? cn2[n] : 0.0f;
        #pragma unroll
        for (int v = 0; v < 8; ++v) {
            float d2 = fn2v[v] + c2 - 2.0f * acc[v];
            d2 = fmaxf(d2, 0.0f);
            const unsigned u = __float_as_uint(d2);
            const int lrow = v + 8 * (lane >> 4);
            if (MODE == 0) {
                if (valid) atomicAdd(&shist[lrow * HBINS + (u >> 21)], 1u);
            } else if (MODE == 1) {
                if (valid && (u >> 21) == b1v[v])
                    atomicAdd(&shist[lrow * HBINS + ((u >> 11) & 1023u)], 1u);
            } else {
                if (valid && u <= Tv[v]) {
                    const float w = __expf(weight[n] - d2 * GCONST);
                    const int lab = labels[n];
                    if ((unsigned)lab < (unsigned)NCLS)
                        atomicAdd(&scls[lrow * NCLS + lab], w);
                }
            }
        }
    }

    __syncthreads();
    // merge LDS accumulators to global (rows rb..rb+15 are contiguous)
    if (MODE == 2) {
        for (int i = tid; i < auxN; i += GTHREADS) {
            const float v = scls[i];
            if (v != 0.0f) atomicAdd(&pacc[(size_t)rb * NCLS + i], v);
        }
    } else {
        for (int i = tid; i < auxN; i += GTHREADS) {
            const unsigned v = shist[i];
            if (v) atomicAdd(&ghist[(size_t)rb * HBINS + i], v);
        }
    }
}

// ---------------- radix-select scans: one wave per row ----------------
__global__ void scan1_kernel(const unsigned* __restrict__ ghist,
                             unsigned* __restrict__ binfo) {
    const int row  = blockIdx.x * (blockDim.x >> 5) + (threadIdx.x >> 5);
    const int lane = threadIdx.x & 31;
    if (row >= BROWS) return;
    const unsigned* h = ghist + (size_t)row * HBINS;
    unsigned s = 0;
    for (int j = 0; j < 32; ++j) s += h[lane * 32 + j];
    unsigned incl = s;
    for (int off = 1; off < 32; off <<= 1) {
        unsigned t = __shfl_up(incl, off, 32);
        if (lane >= off) incl += t;
    }
    const unsigned excl = incl - s;
    const bool hit = (excl < KNN) && (incl >= KNN);
    const unsigned long long msk = __ballot(hit);
    const int hl = (int)__ffsll(msk) - 1;
    if (lane == hl) {
        unsigned c = excl;
        int b = hl * 32;
        while (b < hl * 32 + 31 && c + h[b] < KNN) { c += h[b]; ++b; }
        binfo[row * 2 + 0] = (unsigned)b;   // boundary bin (bits 31..21 key)
        binfo[row * 2 + 1] = c;             // count strictly below it
    }
}

__global__ void scan2_kernel(const unsigned* __restrict__ ghist,
                             const unsigned* __restrict__ binfo,
                             unsigned* __restrict__ thresh) {
    const int row  = blockIdx.x * (blockDim.x >> 5) + (threadIdx.x >> 5);
    const int lane = threadIdx.x & 31;
    if (row >= BROWS) return;
    const unsigned* h  = ghist + (size_t)row * HBINS;
    const unsigned b1  = binfo[row * 2 + 0];
    const unsigned cb  = binfo[row * 2 + 1];
    const unsigned need = KNN - cb;         // >= 1 by construction
    unsigned s = 0;
    for (int j = 0; j < 32; ++j) s += h[lane * 32 + j];
    unsigned incl = s;
    for (int off = 1; off < 32; off <<= 1) {
        unsigned t = __shfl_up(incl, off, 32);
        if (lane >= off) incl += t;
    }
    const unsigned excl = incl - s;
    const bool hit = (excl < need) && (incl >= need);
    const unsigned long long msk = __ballot(hit);
    const int hl = (int)__ffsll(msk) - 1;
    if (lane == hl) {
        unsigned c = excl;
        int b = hl * 32;
        while (b < hl * 32 + 31 && c + h[b] < need) { c += h[b]; ++b; }
        // inclusive threshold: whole 21-bit-prefix sub-bin of the 200th element
        thresh[row] = (b1 << 21) | ((unsigned)b << 11) | 0x7FFu;
    }
}

// ---------------- normalize + log, one row per block ----------------
__global__ void finalize_kernel(float* __restrict__ out) {
    const int row = blockIdx.x;
    float* p  = out + (size_t)BROWS * NCLS + (size_t)row * NCLS;
    float* lg = out + (size_t)row * NCLS;
    __shared__ float red[128];
    float s = 0.0f;
    for (int c = threadIdx.x; c < NCLS; c += 128) {
        float v = p[c];
        if (v == 0.0f) v = 1e-10f;
        p[c] = v;
        s += v;
    }
    red[threadIdx.x] = s;
    __syncthreads();
    for (int off = 64; off > 0; off >>= 1) {
        if (threadIdx.x < off) red[threadIdx.x] += red[threadIdx.x + off];
        __syncthreads();
    }
    const float inv = 1.0f / red[0];
    for (int c = threadIdx.x; c < NCLS; c += 128) {
        const float v = p[c] * inv;
        p[c]  = v;
        lg[c] = logf(v);
    }
}

// ---------------- host-side orchestration ----------------
extern "C" void kernel_launch(void* const* d_in, const int* in_sizes, int n_in,
                              void* d_out, int out_size, void* d_ws, size_t ws_size,
                              hipStream_t stream) {
    (void)in_sizes; (void)n_in; (void)out_size; (void)ws_size;
    const float* feats   = (const float*)d_in[0];   // [2048,128] f32
    const float* centres = (const float*)d_in[1];   // [100000,128] f32
    const float* weight  = (const float*)d_in[2];   // [100000] f32
    const int*   labels  = (const int*)d_in[3];     // [100000] int

    float* out  = (float*)d_out;                    // [0..B*C) logp, [B*C..2B*C) p
    float* pacc = out + (size_t)BROWS * NCLS;

    // workspace layout (float/u32 elements)
    float*    wsf    = (float*)d_ws;
    float*    cn2    = wsf;                         // 100000
    float*    fn2    = wsf + 100352;                // 2048
    unsigned* binfo  = (unsigned*)(wsf + 102400);   // 2048*2
    unsigned* thresh = (unsigned*)(wsf + 106496);   // 2048
    unsigned* ghist  = (unsigned*)(wsf + 108544);   // 2048*1024

    sqnorm_kernel<<<dim3((NCENT + 3) / 4), dim3(128), 0, stream>>>(centres, cn2, NCENT);
    sqnorm_kernel<<<dim3((BROWS + 3) / 4), dim3(128), 0, stream>>>(feats, fn2, BROWS);
    zero_kernel<<<dim3(2048), dim3(256), 0, stream>>>(ghist, (size_t)BROWS * HBINS);

    const int tiles    = (NCENT + TILE_N - 1) / TILE_N;
    const int perSplit = (tiles + CSPLIT - 1) / CSPLIT;
    const int colChunk = perSplit * TILE_N;
    const size_t smemHist = (size_t)(2 * TILE_FLOATS) * 4 + (size_t)16 * HBINS * 4;
    const size_t smemAcc  = (size_t)(2 * TILE_FLOATS) * 4 + (size_t)16 * NCLS * 4;
    const dim3 ggrid(BROWS / 16, CSPLIT);

    gemm_pass_kernel<0><<<ggrid, GTHREADS, smemHist, stream>>>(
        feats, centres, fn2, cn2, ghist, nullptr, nullptr, nullptr, nullptr, nullptr, colChunk);
    scan1_kernel<<<dim3(BROWS / 4), 128, 0, stream>>>(ghist, binfo);
    zero_kernel<<<dim3(2048), dim3(256), 0, stream>>>(ghist, (size_t)BROWS * HBINS);
    gemm_pass_kernel<1><<<ggrid, GTHREADS, smemHist, stream>>>(
        feats, centres, fn2, cn2, ghist, binfo, nullptr, nullptr, nullptr, nullptr, colChunk);
    scan2_kernel<<<dim3(BROWS / 4), 128, 0, stream>>>(ghist, binfo, thresh);
    zero_kernel<<<dim3(2048), dim3(256), 0, stream>>>((unsigned*)pacc, (size_t)BROWS * NCLS);
    gemm_pass_kernel<2><<<ggrid, GTHREADS, smemAcc, stream>>>(
        feats, centres, fn2, cn2, nullptr, nullptr, thresh, weight, labels, pacc, colChunk);
    finalize_kernel<<<dim3(BROWS), 128, 0, stream>>>(out);
}